// GraphSAGEBinary_36507222016452
// MI455X (gfx1250) — compile-verified
//
#include <hip/hip_runtime.h>

// GraphSAGE 2-layer forward for MI455X (gfx1250, wave32).
// Strategy: atomic-scatter mean aggregation (lives in 192MB L2), fused
// f32 WMMA (16x16x4) for the  mean@Wl + h@Wr + b  GEMMs, ReLU fused.

typedef float v2f __attribute__((ext_vector_type(2)));
typedef float v8f __attribute__((ext_vector_type(8)));

#define DIM 64

// ---------------------------------------------------------------------------
// Scatter: for each edge (src -> dst), agg[dst] += feat[src]; optionally count
// degree. 16 threads per edge, float4 gather (coalesced 256B row read),
// 4x global_atomic_add_f32 each (no-return atomics -> STOREcnt path).
// ---------------------------------------------------------------------------
__global__ void sage_scatter_kernel(const float* __restrict__ feat,
                                    const int*   __restrict__ edge,  // [2,E]
                                    float*       __restrict__ agg,   // [N,64]
                                    float*       __restrict__ deg,   // [N]
                                    int addDeg, int nEdges)
{
    long gid = (long)blockIdx.x * blockDim.x + threadIdx.x;
    long e   = gid >> 4;          // edge id
    int  q   = (int)(gid & 15);   // which float4 of the 64-float row
    if (e >= nEdges) return;
    int src = edge[e];
    int dst = edge[(long)nEdges + e];
    const float4 v = ((const float4*)(feat + (long)src * DIM))[q];
    float* a = agg + (long)dst * DIM + q * 4;
    atomicAdd(a + 0, v.x);
    atomicAdd(a + 1, v.y);
    atomicAdd(a + 2, v.z);
    atomicAdd(a + 3, v.w);
    if (addDeg && q == 0) atomicAdd(deg + dst, 1.0f);
}

// ---------------------------------------------------------------------------
// Fused SAGEConv GEMM:  out = relu( (agg/deg) @ wl + xin @ wr + bias )
// One wave per 16x16 output tile. 8 waves/block -> 32 rows x 64 cols per block.
// A fragment (16x4 f32): lanes 0-15 hold row M=lane, K=k{+0,+1};
//                        lanes 16-31 hold row M=lane-16, K=k{+2,+3}.
// B fragment (4x16 f32): lanes 0-15 hold col N=lane, K=k{+0,+1};
//                        lanes 16-31 hold col N=lane-16, K=k{+2,+3}.
// C/D (16x16 f32): VGPR r -> M=r (lanes 0-15) / M=r+8 (lanes 16-31), N=lane&15.
// ---------------------------------------------------------------------------
__global__ void sage_gemm_kernel(const float* __restrict__ agg,   // [N,64] sums
                                 const float* __restrict__ deg,   // [N]
                                 const float* __restrict__ xin,   // [N,64] root feats
                                 const float* __restrict__ wl,    // [64,64]
                                 const float* __restrict__ wr,    // [64,64]
                                 const float* __restrict__ bias,  // [64]
                                 float*       __restrict__ out,   // [N,64]
                                 int nNodes)
{
    const int lane    = threadIdx.x & 31;
    const int wave    = threadIdx.x >> 5;
    const int rowTile = blockIdx.x * 2 + (wave >> 2);  // 2 row-tiles per block
    const int colTile = wave & 3;                      // 4 col-tiles (64 cols)
    const int row0    = rowTile << 4;
    if (row0 >= nNodes) return;                        // wave-uniform guard

    const int lrow = row0 + (lane & 15);               // A row for this lane
    const int kb   = (lane >> 4) << 1;                 // K sub-offset: 0 or 2
    const int col  = (colTile << 4) + (lane & 15);     // B/C column

    const float invd = 1.0f / fmaxf(deg[lrow], 1.0f);
    const float* arow = agg + (long)lrow * DIM;
    const float* xrow = xin + (long)lrow * DIM;

    v8f acc;
    const float bcol = bias[col];
#pragma unroll
    for (int r = 0; r < 8; ++r) acc[r] = bcol;

#pragma unroll
    for (int k = 0; k < DIM; k += 4) {
        const int ka = k + kb;
        v2f a, b;
        // mean-aggregated term: (agg/deg) @ wl
        a.x = arow[ka]     * invd;
        a.y = arow[ka + 1] * invd;
        b.x = wl[ka * DIM + col];
        b.y = wl[(ka + 1) * DIM + col];
        acc = __builtin_amdgcn_wmma_f32_16x16x4_f32(false, a, false, b,
                                                    (short)0, acc, false, false);
        // root term: xin @ wr
        a.x = xrow[ka];
        a.y = xrow[ka + 1];
        b.x = wr[ka * DIM + col];
        b.y = wr[(ka + 1) * DIM + col];
        acc = __builtin_amdgcn_wmma_f32_16x16x4_f32(false, a, false, b,
                                                    (short)0, acc, false, false);
    }

    const int rbase = row0 + ((lane >> 4) << 3);
#pragma unroll
    for (int r = 0; r < 8; ++r) {
        float v = acc[r];
        v = v > 0.0f ? v : 0.0f;                       // fused ReLU
        out[(long)(rbase + r) * DIM + col] = v;
    }
}

// ---------------------------------------------------------------------------
// Output head: out[i] = dot(h[i,:], w3) + b3
// ---------------------------------------------------------------------------
__global__ void sage_head_kernel(const float* __restrict__ h,   // [N,64]
                                 const float* __restrict__ w3,  // [64]
                                 const float* __restrict__ b3,  // [1]
                                 float*       __restrict__ out, // [N]
                                 int n)
{
    int i = blockIdx.x * blockDim.x + threadIdx.x;
    if (i >= n) return;
    const float4* hp = (const float4*)(h + (long)i * DIM);
    const float4* wp = (const float4*)w3;
    float s = 0.0f;
#pragma unroll
    for (int q = 0; q < 16; ++q) {
        float4 a = hp[q];
        float4 w = wp[q];
        s += a.x * w.x + a.y * w.y + a.z * w.z + a.w * w.w;
    }
    out[i] = s + b3[0];
}

// ---------------------------------------------------------------------------
extern "C" void kernel_launch(void* const* d_in, const int* in_sizes, int n_in,
                              void* d_out, int out_size, void* d_ws, size_t ws_size,
                              hipStream_t stream)
{
    const float* x   = (const float*)d_in[0];   // [N,64]
    const int*   edg = (const int*)  d_in[1];   // [2,E]
    const float* w1l = (const float*)d_in[2];
    const float* w1r = (const float*)d_in[3];
    const float* b1  = (const float*)d_in[4];
    const float* w2l = (const float*)d_in[5];
    const float* w2r = (const float*)d_in[6];
    const float* b2  = (const float*)d_in[7];
    const float* w3  = (const float*)d_in[8];
    const float* b3  = (const float*)d_in[9];
    float* out = (float*)d_out;

    const int n = in_sizes[0] / DIM;   // 100000
    const int e = in_sizes[1] / 2;     // 1600000

    // Workspace layout: [deg: N f32][agg: N*64 f32][h1: N*64 f32][h2: N*64 f32]
    char* ws = (char*)d_ws;
    const size_t degBytes  = (size_t)n * sizeof(float);
    const size_t featBytes = (size_t)n * DIM * sizeof(float);
    float* deg = (float*)ws;
    float* agg = (float*)(ws + degBytes);
    float* h1  = (float*)(ws + degBytes + featBytes);
    float* h2  = (float*)(ws + degBytes + 2 * featBytes);

    const int scatterBlocks = (e * 16 + 255) / 256;
    const int gemmBlocks    = (n + 31) / 32;   // 32 rows per block (8 waves)

    // ---- Layer 1 ----
    hipMemsetAsync(ws, 0, degBytes + featBytes, stream);          // deg + agg
    sage_scatter_kernel<<<scatterBlocks, 256, 0, stream>>>(x, edg, agg, deg, 1, e);
    sage_gemm_kernel<<<gemmBlocks, 256, 0, stream>>>(agg, deg, x, w1l, w1r, b1, h1, n);

    // ---- Layer 2 (degree unchanged; re-zero agg only) ----
    hipMemsetAsync(agg, 0, featBytes, stream);
    sage_scatter_kernel<<<scatterBlocks, 256, 0, stream>>>(h1, edg, agg, deg, 0, e);
    sage_gemm_kernel<<<gemmBlocks, 256, 0, stream>>>(agg, deg, h1, w2l, w2r, b2, h2, n);

    // ---- Head ----
    sage_head_kernel<<<(n + 255) / 256, 256, 0, stream>>>(h2, w3, b3, out, n);
}